// PPNPGuard_9363028706291
// MI455X (gfx1250) — compile-verified
//
#include <hip/hip_runtime.h>
#include <hip/hip_bf16.h>
#include <math.h>

// ---------------- problem constants ----------------
#define NNODE   4096
#define NEDGE   131072
#define DIN     512
#define DHID    256
#define DOUT    32
#define ALPHA   0.85f
#define ONEMA   0.15f
#define EPSF    1e-8f
#define NITER   128      // 0.85^128 ~ 1e-9: below fp32 noise

typedef __attribute__((ext_vector_type(2))) float v2f;
typedef __attribute__((ext_vector_type(8))) float v8f;

__device__ __forceinline__ v8f wmma_f32_16x16x4(v2f a, v2f b, v8f c) {
  // (neg_a, A, neg_b, B, c_mod, C, reuse_a, reuse_b)
  return __builtin_amdgcn_wmma_f32_16x16x4_f32(false, a, false, b, (short)0, c,
                                               false, false);
}

// ---------------- utility kernels ----------------
__global__ void zero_f4_kernel(float4* __restrict__ p, size_t n4) {
  size_t i = (size_t)blockIdx.x * blockDim.x + threadIdx.x;
  size_t stride = (size_t)gridDim.x * blockDim.x;
  float4 z; z.x = z.y = z.z = z.w = 0.0f;
  for (; i < n4; i += stride) p[i] = z;
}

__global__ void scale_copy_kernel(const float* __restrict__ in,
                                  float* __restrict__ out, float s, int n) {
  int i = blockIdx.x * blockDim.x + threadIdx.x;
  if (i < n) out[i] = s * in[i];
}

// ---------------- GEMM: C = act(A[MxK] @ B[KxN] + bias) via fp32 WMMA ----------------
// one wave per 16x16 output tile; K stepped by 4 through v_wmma_f32_16x16x4_f32
__global__ void gemm_wmma_kernel(const float* __restrict__ A,
                                 const float* __restrict__ B,
                                 const float* __restrict__ bias,
                                 float* __restrict__ C,
                                 int M, int K, int N, int relu) {
  int wave = blockIdx.x * (blockDim.x >> 5) + (threadIdx.x >> 5);
  int ntiles = N >> 4;
  int total  = (M >> 4) * ntiles;
  if (wave >= total) return;
  int mt = wave / ntiles, nt = wave - mt * ntiles;
  int m0 = mt << 4, n0 = nt << 4;
  int lane = threadIdx.x & 31;
  int l15  = lane & 15;
  int half = lane >> 4;       // 0 or 1
  int kb   = half << 1;       // K sub-offset 0 or 2

  const float* arow = A + (size_t)(m0 + l15) * K;
  v8f c = {};
  for (int k = 0; k < K; k += 4) {
    v2f a, b;
    a.x = arow[k + kb];
    a.y = arow[k + kb + 1];
    b.x = B[(size_t)(k + kb) * N + n0 + l15];
    b.y = B[(size_t)(k + kb + 1) * N + n0 + l15];
    c = wmma_f32_16x16x4(a, b, c);
    __builtin_prefetch(&arow[k + 32], 0, 1);   // global_prefetch_b8
  }
  float bv = bias ? bias[n0 + l15] : 0.0f;
  int mbase = m0 + half * 8;
#pragma unroll
  for (int r = 0; r < 8; ++r) {
    float v = c[r] + bv;
    if (relu) v = fmaxf(v, 0.0f);
    C[(size_t)(mbase + r) * N + n0 + l15] = v;
  }
}

// ---------------- row-normalize logits: F = L / max(||L_i||, eps) ----------------
__global__ void rownorm_kernel(const float* __restrict__ L,
                               float* __restrict__ F, int n) {
  int row  = blockIdx.x * (blockDim.x >> 5) + (threadIdx.x >> 5);
  int lane = threadIdx.x & 31;
  if (row >= n) return;
  float v = L[row * DOUT + lane];
  float s = v * v;
#pragma unroll
  for (int off = 16; off; off >>= 1) s += __shfl_xor(s, off, 32);
  float nrm = fmaxf(sqrtf(s), EPSF);
  F[row * DOUT + lane] = v / nrm;
}

// ---------------- per-edge cosine similarity + threshold ----------------
__global__ void edgesim_kernel(const float* __restrict__ F,
                               const int* __restrict__ row,
                               const int* __restrict__ col,
                               float* __restrict__ sim, int E) {
  int e    = blockIdx.x * (blockDim.x >> 5) + (threadIdx.x >> 5);
  int lane = threadIdx.x & 31;
  if (e >= E) return;
  int r = row[e], cc = col[e];
  float p = F[r * DOUT + lane] * F[cc * DOUT + lane];
#pragma unroll
  for (int off = 16; off; off >>= 1) p += __shfl_xor(p, off, 32);
  if (lane == 0) sim[e] = (p < 0.1f) ? 0.0f : p;
}

// ---------------- scatter sim into dense S (set semantics; dup -> one wins) ----------------
__global__ void scatter_kernel(const int* __restrict__ row,
                               const int* __restrict__ col,
                               const float* __restrict__ sim,
                               float* __restrict__ S, int E) {
  int e = blockIdx.x * blockDim.x + threadIdx.x;
  if (e < E) S[(size_t)row[e] * NNODE + col[e]] = sim[e];
}

// ---------------- global scalar flags from S[0,0] (before S is rewritten) ----------------
__global__ void flags_kernel(const float* __restrict__ S, float* __restrict__ flags) {
  float s00 = S[0];
  bool removeDiag = (s00 == 1.0f);
  float s00p = removeDiag ? 0.0f : s00;
  flags[0] = removeDiag ? 1.0f : 0.0f;
  flags[1] = (s00p == 0.0f) ? 1.0f : 0.0f;   // addDiag
}

// ---------------- per-row: S -> alpha * D^{-1} A, in place (one block per row) ----
__global__ void rowproc_kernel(float* __restrict__ S, const float* __restrict__ flags) {
  __shared__ float redf[256];
  __shared__ int   redi[256];
  int i = blockIdx.x;
  int t = threadIdx.x;
  bool removeDiag = flags[0] != 0.0f;
  bool addDiag    = flags[1] != 0.0f;
  float* rowp = S + (size_t)i * NNODE;

  // sweep 1: L1 row sum (diag removed if requested) + nonzero count
  float asum = 0.0f; int cnt = 0;
  for (int j = t; j < NNODE; j += 256) {
    float v = rowp[j];
    if (removeDiag && j == i) v = 0.0f;
    asum += fabsf(v);
    cnt  += (v != 0.0f) ? 1 : 0;
  }
  redf[t] = asum; redi[t] = cnt;
  __syncthreads();
  for (int s = 128; s > 0; s >>= 1) {
    if (t < s) { redf[t] += redf[t + s]; redi[t] += redi[t + s]; }
    __syncthreads();
  }
  float rowsum = redf[0];
  int   deg    = redi[0];
  float rdiv   = (rowsum == 0.0f) ? 1.0f : rowsum;
  float lam    = 1.0f / ((float)deg + 1.0f);
  __syncthreads();

  // sweep 2: Sn -> A = exp over sparsity pattern, write in place, accumulate deg(A)
  float dsum = 0.0f;
  for (int j = t; j < NNODE; j += 256) {
    float v = rowp[j];
    if (removeDiag && j == i) v = 0.0f;
    float sn = v / rdiv;
    if (addDiag && j == i) sn += lam;
    float a = (sn != 0.0f) ? expf(sn) : 0.0f;
    rowp[j] = a;
    dsum += a;
  }
  redf[t] = dsum;
  __syncthreads();
  for (int s = 128; s > 0; s >>= 1) {
    if (t < s) redf[t] += redf[t + s];
    __syncthreads();
  }
  float degA  = redf[0];
  float scale = ALPHA / fmaxf(degA, EPSF);
  __syncthreads();

  // sweep 3: row of alpha * P
  for (int j = t; j < NNODE; j += 256) rowp[j] *= scale;
}

// ---------------- power-iteration step: Zout = (alphaP) @ Zin + 0.15*L ----------------
// block = 512 thr = 16 waves = 4 M-tiles x 2 N-tiles x 2 K-splits.
// Zin K-chunks staged once per half-block into LDS (transposed, padded pitch 66
// -> conflict-free and 8B-aligned so B-operand pair is one ds_load_b64).
// Traffic per iteration: P 64MB (irreducible) + Zin 32MB, all L2-resident.
__global__ void prop_kernel(const float* __restrict__ P,
                            const float* __restrict__ Zin,
                            const float* __restrict__ L,
                            float* __restrict__ Zout) {
  __shared__ float Zt[2][32][66];   // [ksplit][col][k-in-chunk], ~16.5 KB
  __shared__ float acc[8][32][8];   // ks==1 partials, 8 KB
  int tid  = threadIdx.x;
  int wave = tid >> 5, lane = tid & 31;
  int mt = wave & 3;            // M sub-tile 0..3
  int nt = (wave >> 2) & 1;     // N tile
  int ks = wave >> 3;           // K split 0/1 (2048 each)
  int m0 = blockIdx.x * 64 + mt * 16;
  int n0 = nt << 4;
  int l15 = lane & 15, half = lane >> 4, kb = half << 1;
  int kBase = ks * 2048;
  int hid = tid & 255;          // id within the 8-wave half-block

  const float* prow = P + (size_t)(m0 + l15) * NNODE + kBase;
  v8f c = {};
  for (int kc = 0; kc < 2048; kc += 64) {
    __syncthreads();            // previous chunk fully consumed
    // stage Zin[kBase+kc .. +64) transposed; coalesced global reads
    for (int u = hid; u < 64 * 32; u += 256) {
      int r  = u >> 5;          // k within chunk
      int cc = u & 31;          // column
      Zt[ks][cc][r] = Zin[(kBase + kc + r) * DOUT + cc];
    }
    __syncthreads();
#pragma unroll
    for (int kk = 0; kk < 64; kk += 4) {
      v2f a, b;
      a.x = prow[kc + kk + kb];
      a.y = prow[kc + kk + kb + 1];
      b.x = Zt[ks][n0 + l15][kk + kb];      // contiguous pair -> ds_load_b64
      b.y = Zt[ks][n0 + l15][kk + kb + 1];
      c = wmma_f32_16x16x4(a, b, c);
    }
    __builtin_prefetch(&prow[kc + 128], 0, 1);  // stream next P chunk from L2
  }
  // reduce the two K-splits
  if (ks == 1) {
#pragma unroll
    for (int r = 0; r < 8; ++r) acc[wave - 8][lane][r] = c[r];
  }
  __syncthreads();
  if (ks == 0) {
#pragma unroll
    for (int r = 0; r < 8; ++r) {
      float s = c[r] + acc[wave][lane][r];
      int m = m0 + half * 8 + r;
      Zout[m * DOUT + n0 + l15] = s + ONEMA * L[m * DOUT + n0 + l15];
    }
  }
}

// ---------------- host orchestration ----------------
extern "C" void kernel_launch(void* const* d_in, const int* in_sizes, int n_in,
                              void* d_out, int out_size, void* d_ws, size_t ws_size,
                              hipStream_t stream) {
  const float* attr = (const float*)d_in[0];   // [4096, 512]
  const int*   row  = (const int*)d_in[1];     // [131072]
  const int*   col  = (const int*)d_in[2];     // [131072]
  const float* W0   = (const float*)d_in[3];   // [512, 256]
  const float* b0   = (const float*)d_in[4];   // [256]
  const float* W1   = (const float*)d_in[5];   // [256, 32]
  const float* b1   = (const float*)d_in[6];   // [32]
  float* out = (float*)d_out;                  // [4096, 32]

  // workspace layout (floats)
  float* S     = (float*)d_ws;                       // 4096*4096 (becomes alpha*P)
  float* H     = S + (size_t)NNODE * NNODE;          // 4096*256
  float* Lg    = H + (size_t)NNODE * DHID;           // 4096*32 logits
  float* F     = Lg + (size_t)NNODE * DOUT;          // 4096*32
  float* sim   = F + (size_t)NNODE * DOUT;           // 131072
  float* flags = sim + NEDGE;                        // 2 (+pad)
  float* Za    = flags + 16;                         // 4096*32
  float* Zb    = Za + (size_t)NNODE * DOUT;          // 4096*32

  // 1) zero dense S
  zero_f4_kernel<<<4096, 256, 0, stream>>>((float4*)S,
                                           ((size_t)NNODE * NNODE) / 4);

  // 2) H = relu(attr @ W0 + b0)   [4096x512x256], 4096 wmma tiles
  gemm_wmma_kernel<<<1024, 128, 0, stream>>>(attr, W0, b0, H,
                                             NNODE, DIN, DHID, 1);

  // 3) logits = H @ W1 + b1       [4096x256x32], 512 tiles
  gemm_wmma_kernel<<<128, 128, 0, stream>>>(H, W1, b1, Lg,
                                            NNODE, DHID, DOUT, 0);

  // 4) F = row-normalized logits
  rownorm_kernel<<<NNODE / 8, 256, 0, stream>>>(Lg, F, NNODE);

  // 5) edge-wise cosine similarity with threshold
  edgesim_kernel<<<NEDGE / 8, 256, 0, stream>>>(F, row, col, sim, NEDGE);

  // 6) scatter into dense S
  scatter_kernel<<<NEDGE / 256, 256, 0, stream>>>(row, col, sim, S, NEDGE);

  // 7) global flags from S[0,0] (must precede in-place rewrite of S)
  flags_kernel<<<1, 1, 0, stream>>>(S, flags);

  // 8) S -> alpha * D^{-1} A  (in place, one block per row)
  rowproc_kernel<<<NNODE, 256, 0, stream>>>(S, flags);

  // 9) Z0 = (1 - alpha) * logits
  scale_copy_kernel<<<(NNODE * DOUT) / 256, 256, 0, stream>>>(Lg, Za, ONEMA,
                                                              NNODE * DOUT);

  // 10) power iteration: Z <- alphaP @ Z + (1-alpha) L ; final step -> d_out
  float* cur = Za;
  float* nxt = Zb;
  for (int it = 0; it < NITER; ++it) {
    float* o = (it == NITER - 1) ? out : nxt;
    prop_kernel<<<NNODE / 64, 512, 0, stream>>>(S, cur, Lg, o);
    float* t = cur; cur = o; nxt = t;
  }
}